// IntrinsicMotivationManager_37082747634613
// MI455X (gfx1250) — compile-verified
//
#include <hip/hip_runtime.h>
#include <hip/hip_bf16.h>

typedef __attribute__((ext_vector_type(2))) float v2f;
typedef __attribute__((ext_vector_type(8))) float v8f;

#define N_ROWS 16384
#define FEAT   2048
#define BINS   32
#define KSTEPS (FEAT / 4)     // 512 WMMA K-steps (K=4 each)
#define KC     128            // K-steps per LDS chunk (64KB)
#define NCHUNK (KSTEPS / KC)  // 4

// ---------------- K1: partial column sums / sums of squares ----------------
// grid (8, 64), block 256. Coalesced: consecutive lanes read consecutive cols.
__global__ void k1_colstats(const float* __restrict__ X,
                            float* __restrict__ psum, float* __restrict__ psq) {
  const int k = blockIdx.x * 256 + threadIdx.x;  // column 0..2047
  const int rbase = blockIdx.y * 256;            // row chunk
  float s = 0.f, q = 0.f;
  for (int r = 0; r < 256; ++r) {
    float v = X[(size_t)(rbase + r) * FEAT + k];
    s += v; q += v * v;
  }
  psum[blockIdx.y * FEAT + k] = s;
  psq [blockIdx.y * FEAT + k] = q;
}

// ---------------- K2: finalize RunningMeanStd mean / 1/sigma ----------------
__global__ void k2_finalize(const float* __restrict__ psum, const float* __restrict__ psq,
                            float* __restrict__ meanv, float* __restrict__ isig) {
  const int k = blockIdx.x * 256 + threadIdx.x;
  float s = 0.f, q = 0.f;
  for (int rc = 0; rc < 64; ++rc) { s += psum[rc * FEAT + k]; q += psq[rc * FEAT + k]; }
  const float n  = (float)N_ROWS;
  const float bm = s / n;
  const float bv = (q - n * bm * bm) / (n - 1.0f);   // unbiased var
  const float eps = 1e-4f;                            // RMS initial count
  const float tot = eps + n;
  const float mean = bm * (n / tot);
  const float m2   = eps + bv * n + bm * bm * (eps * n / tot);
  const float var  = m2 / tot;
  const float sigma = sqrtf(var + 1e-8f);
  meanv[k] = mean;
  isig [k] = 1.0f / sigma;
}

// ---------------- K3a: pack B = P * (1/sigma) into WMMA lane layout ----------------
// B fragment (4x16 f32, 2 VGPRs): VGPR v, lane<16 -> K=v,   N=lane
//                                          lane>=16 -> K=v+2, N=lane-16
// Flat layout: idx = s*128 + tile*64 + lane*2 + v   (s = K-step)
__global__ void k3_pack(const float* __restrict__ P, const float* __restrict__ isig,
                        float* __restrict__ bpack) {
  const int tid  = blockIdx.x * 256 + threadIdx.x;   // 0..65535
  const int v    = tid & 1;
  const int lane = (tid >> 1) & 31;
  const int t    = (tid >> 6) & 1;
  const int s    = tid >> 7;
  const int kb   = v + ((lane < 16) ? 0 : 2);
  const int k    = 4 * s + kb;
  const int j    = t * 16 + (lane & 15);
  bpack[tid] = P[k * BINS + j] * isig[k];
}

// ---------------- K3b: bias[j] = sum_k mean[k]*isig[k]*P[k][j] ----------------
__global__ void k3_bias(const float* __restrict__ P, const float* __restrict__ meanv,
                        const float* __restrict__ isig, float* __restrict__ bias) {
  const int j = threadIdx.x;  // 0..31, one wave
  float acc = 0.f;
  for (int k = 0; k < FEAT; ++k) acc += meanv[k] * isig[k] * P[k * BINS + j];
  bias[j] = acc;
}

// ---------------- K4: fp32 WMMA GEMM + ballot-based sign hash ----------------
// 8 waves/block, each wave owns 16 rows; 128 blocks -> 16384 rows.
// B staged in 64KB LDS chunks; signs -> 32-bit hash via two ballots per 2 rows.
__global__ void __launch_bounds__(256) k4_wmma_hash(const float* __restrict__ X,
                                                    const float* __restrict__ bpack,
                                                    const float* __restrict__ bias,
                                                    unsigned long long* __restrict__ keys) {
  __shared__ float lds[KC * 128];  // 16384 floats = 64KB of the 320KB WGP LDS
  const int tid  = threadIdx.x;
  const int lane = tid & 31;
  const int wave = tid >> 5;
  const int row0 = (blockIdx.x * 8 + wave) * 16;
  const int rowA = row0 + (lane & 15);
  const int ksb  = (lane < 16) ? 0 : 2;  // A frag: lanes 0-15 hold K=0,1; 16-31 hold K=2,3
  const float* xrow = X + (size_t)rowA * FEAT + ksb;

  v8f c0 = {0.f,0.f,0.f,0.f,0.f,0.f,0.f,0.f};  // cols 0..15
  v8f c1 = {0.f,0.f,0.f,0.f,0.f,0.f,0.f,0.f};  // cols 16..31

  for (int c = 0; c < NCHUNK; ++c) {
    // cooperative 64KB chunk load, 16 x float4 per thread
    const float4* src = (const float4*)(bpack + c * (KC * 128));
    float4* dst = (float4*)lds;
#pragma unroll
    for (int i = 0; i < 16; ++i) dst[tid + i * 256] = src[tid + i * 256];
    __syncthreads();

    const int sbase = c * KC;
#pragma unroll 4
    for (int sl = 0; sl < KC; ++sl) {
      const int k0 = 4 * (sbase + sl);
      v2f a  = *(const v2f*)(xrow + k0);
      v2f b0 = *(const v2f*)(lds + sl * 128 +  0 + lane * 2);
      v2f b1 = *(const v2f*)(lds + sl * 128 + 64 + lane * 2);
      c0 = __builtin_amdgcn_wmma_f32_16x16x4_f32(false, a, false, b0, (short)0, c0, false, false);
      c1 = __builtin_amdgcn_wmma_f32_16x16x4_f32(false, a, false, b1, (short)0, c1, false, false);
    }
    __syncthreads();
  }

  // C/D layout: VGPR v, lane<16 -> M=v,N=lane ; lane>=16 -> M=v+8,N=lane-16
  const float bj0 = bias[lane & 15];
  const float bj1 = bias[(lane & 15) + 16];
#pragma unroll
  for (int v = 0; v < 8; ++v) {
    unsigned m0 = __builtin_amdgcn_ballot_w32(c0[v] > bj0);  // cols 0-15, rows v / v+8
    unsigned m1 = __builtin_amdgcn_ballot_w32(c1[v] > bj1);  // cols 16-31
    unsigned hA = (m0 & 0xFFFFu) | ((m1 & 0xFFFFu) << 16);   // row0+v
    unsigned hB = (m0 >> 16)     | (m1 & 0xFFFF0000u);       // row0+v+8
    if (lane == 0) {
      const int rA = row0 + v, rB = row0 + v + 8;
      keys[rA] = ((unsigned long long)hA << 6) | (unsigned)(rA & 63);  // h*64 + env
      keys[rB] = ((unsigned long long)hB << 6) | (unsigned)(rB & 63);
    }
  }
}

// ---------------- K5: ordered occurrence rank + reward ----------------
// rank_i = 1 + #{j<i : key_j == key_i}; LDS-tiled triangular scan, deterministic.
__global__ void __launch_bounds__(256) k5_rank(const unsigned long long* __restrict__ keys,
                                               float* __restrict__ out) {
  __shared__ unsigned long long sk[256];
  const int i = blockIdx.x * 256 + threadIdx.x;
  const unsigned long long my = keys[i];
  int cnt = 0;
  for (int tb = 0; tb <= (int)blockIdx.x; ++tb) {
    sk[threadIdx.x] = keys[tb * 256 + threadIdx.x];
    __syncthreads();
    int limit = i - tb * 256; if (limit > 256) limit = 256;
    for (int jj = 0; jj < limit; ++jj) cnt += (sk[jj] == my) ? 1 : 0;
    __syncthreads();
  }
  out[i] = 1.0f / sqrtf((float)(cnt + 1));
}

// ---------------- launch ----------------
extern "C" void kernel_launch(void* const* d_in, const int* in_sizes, int n_in,
                              void* d_out, int out_size, void* d_ws, size_t ws_size,
                              hipStream_t stream) {
  (void)in_sizes; (void)n_in; (void)out_size; (void)ws_size;
  const float* X = (const float*)d_in[0];   // features (64,256,2048) f32
  const float* P = (const float*)d_in[1];   // random_projection (2048,32) f32
  float* out = (float*)d_out;               // rewards, 16384 f32

  char* ws = (char*)d_ws;
  float* psum  = (float*)(ws + 0);          // 64*2048 f32
  float* psq   = (float*)(ws + 524288);     // 64*2048 f32
  float* meanv = (float*)(ws + 1048576);    // 2048 f32
  float* isig  = (float*)(ws + 1056768);    // 2048 f32
  float* bias  = (float*)(ws + 1064960);    // 32 f32
  float* bpack = (float*)(ws + 1065088);    // 65536 f32 (16B aligned)
  unsigned long long* keys = (unsigned long long*)(ws + 1327232);  // 16384 u64

  k1_colstats<<<dim3(8, 64), 256, 0, stream>>>(X, psum, psq);
  k2_finalize<<<8, 256, 0, stream>>>(psum, psq, meanv, isig);
  k3_pack    <<<256, 256, 0, stream>>>(P, isig, bpack);
  k3_bias    <<<1, 32, 0, stream>>>(P, meanv, isig, bias);
  k4_wmma_hash<<<128, 256, 0, stream>>>(X, bpack, bias, keys);
  k5_rank    <<<64, 256, 0, stream>>>(keys, out);
}